// PairProposalNetwork_12824772346133
// MI455X (gfx1250) — compile-verified
//
#include <hip/hip_runtime.h>

// ---------------------------------------------------------------------------
// PairProposalNetwork on MI455X (gfx1250, wave32, WMMA)
//   N=256 pairs, T=256 tokens, D=1024 enc dim, H=1024 hidden
// Pipeline:
//   K1: max-pool over T (HBM-bound, float4 coalesced)       -> sub, obj [N,D]
//   K2: f32 WMMA GEMM  s_part = sub @ W1[:, :D]^T + b1      -> [N,H]
//                      o_part = obj @ W1[:, D:]^T           -> [N,H]
//   K3: fused scoring  out[i,j] = sum_h relu(s[i,h]+o[j,h])*W2[h] + b2
//       (b1 pre-folded into s_part), diagonal forced to 0
// ---------------------------------------------------------------------------

typedef __attribute__((ext_vector_type(2))) float v2f;
typedef __attribute__((ext_vector_type(8))) float v8f;

#define NN 256
#define TT 256
#define DD 1024
#define HH 1024

// ---------------------------------------------------------------- K1: maxpool
// grid (1, N, 2), block 256. Each thread owns 4 consecutive d's (float4),
// loops over T with stride D — fully coalesced, streams 512 MB from HBM.
__global__ void ppn_maxpool_kernel(const float* __restrict__ subs,
                                   const float* __restrict__ objs,
                                   float* __restrict__ outSub,
                                   float* __restrict__ outObj) {
  const int n     = blockIdx.y;
  const int which = blockIdx.z;
  const float* in  = which ? objs   : subs;
  float*       out = which ? outObj : outSub;

  const int d4 = (blockIdx.x * blockDim.x + threadIdx.x) * 4;
  const float* base = in + (size_t)n * TT * DD + d4;

  float4 m = *(const float4*)(base);
#pragma unroll 4
  for (int t = 1; t < TT; ++t) {
    float4 v = *(const float4*)(base + (size_t)t * DD);
    m.x = fmaxf(m.x, v.x);
    m.y = fmaxf(m.y, v.y);
    m.z = fmaxf(m.z, v.z);
    m.w = fmaxf(m.w, v.w);
  }
  *(float4*)(out + (size_t)n * DD + d4) = m;
}

// ------------------------------------------------------------- K2: WMMA GEMM
// C[n,h] = sum_d A[n,d] * W1[h, colOff + d]   (+ b1[h] for the subject path)
// One wave per 16x16 C tile; K-loop over D in steps of 4 using
// V_WMMA_F32_16X16X4_F32.
//
// ISA VGPR layouts (CDNA5 ISA §7.12.2):
//   A 16x4 f32 : lanes 0-15 -> M=lane,    {V0,V1} = K{0,1}
//                lanes16-31 -> M=lane-16, {V0,V1} = K{2,3}
//   B  4x16 f32: symmetric (N=lane%16, K half selected by lane/16)
//   C 16x16 f32: lane -> N=lane%16; VGPR r -> M = r + 8*(lane/16)
__global__ void ppn_gemm_wmma_kernel(const float* __restrict__ sub,   // [N,D]
                                     const float* __restrict__ obj,   // [N,D]
                                     const float* __restrict__ W1,    // [H,2D]
                                     const float* __restrict__ b1,    // [H]
                                     float* __restrict__ sPartB,      // [N,H]
                                     float* __restrict__ oPart) {     // [N,H]
  const int waveId = threadIdx.x >> 5;
  const int lane   = threadIdx.x & 31;
  const int wavesPerBlock = blockDim.x >> 5;

  int tile = blockIdx.x * wavesPerBlock + waveId;
  const int tilesPerMat = (NN / 16) * (HH / 16);      // 16*64 = 1024
  const int sel = tile / tilesPerMat;                  // 0=subject, 1=object
  tile -= sel * tilesPerMat;
  const int ti = tile / (HH / 16);                     // row-tile (N)
  const int th = tile % (HH / 16);                     // col-tile (H)
  const int row0 = ti * 16;
  const int h0   = th * 16;

  const float* A    = sel ? obj : sub;
  const float* Wcol = W1 + (sel ? DD : 0);             // column block of W1

  const int m    = lane & 15;
  const int half = lane >> 4;                          // 0: K=0,1  1: K=2,3

  const float* aRow = A    + (size_t)(row0 + m) * DD;        // A[m, :]
  const float* bRow = Wcol + (size_t)(h0   + m) * (2 * DD);  // W1[h0+n, :]

  v8f c = {0.f, 0.f, 0.f, 0.f, 0.f, 0.f, 0.f, 0.f};

  for (int k0 = 0; k0 < DD; k0 += 4) {
    const int kk = k0 + half * 2;                      // even -> 8B aligned
    v2f a = *(const v2f*)(aRow + kk);
    v2f b = *(const v2f*)(bRow + kk);
    // 8 args: (neg_a, A, neg_b, B, c_mod, C, reuse_a, reuse_b)
    c = __builtin_amdgcn_wmma_f32_16x16x4_f32(
        /*neg_a=*/false, a, /*neg_b=*/false, b,
        /*c_mod=*/(short)0, c, /*reuse_a=*/false, /*reuse_b=*/false);
  }

  const int h = h0 + m;
  const float bias = sel ? 0.0f : b1[h];               // fold b1 into s_part
  float* dst = sel ? oPart : sPartB;
#pragma unroll
  for (int r = 0; r < 8; ++r) {
    const int row = row0 + r + 8 * half;
    dst[(size_t)row * HH + h] = c[r] + bias;
  }
}

// ----------------------------------------------------------- K3: pair scores
// block = 256 threads = 16x16 (j fast, i slow) tile of the [N,N] output.
// h processed in 64-wide chunks staged in LDS (rows padded to 65 floats to
// avoid the stride-64 bank conflict on the o-tile reads).
#define CH 64
__global__ void ppn_score_kernel(const float* __restrict__ sPartB,  // [N,H]
                                 const float* __restrict__ oPart,   // [N,H]
                                 const float* __restrict__ W2,      // [H]
                                 const float* __restrict__ b2,      // [1]
                                 float* __restrict__ out) {         // [N,N]
  __shared__ float sS[16][CH + 1];
  __shared__ float sO[16][CH + 1];
  __shared__ float sW[CH];

  const int tx = threadIdx.x & 15;    // j within tile
  const int ty = threadIdx.x >> 4;    // i within tile
  const int i0 = blockIdx.y * 16;
  const int j0 = blockIdx.x * 16;

  float acc = 0.0f;

  for (int h0 = 0; h0 < HH; h0 += CH) {
    __syncthreads();
    {
      const int r = threadIdx.x / CH;         // 0..3
      const int cc = threadIdx.x % CH;        // 0..63 (contiguous -> coalesced)
#pragma unroll
      for (int rr = 0; rr < 4; ++rr) {
        const int row = r + rr * 4;
        sS[row][cc] = sPartB[(size_t)(i0 + row) * HH + h0 + cc];
        sO[row][cc] = oPart [(size_t)(j0 + row) * HH + h0 + cc];
      }
      if (threadIdx.x < CH) sW[threadIdx.x] = W2[h0 + threadIdx.x];
    }
    __syncthreads();

#pragma unroll 8
    for (int h = 0; h < CH; ++h) {
      // sS[ty][h]: lane-broadcast; sO[tx][h]: conflict-free via +1 pad
      acc = fmaf(fmaxf(sS[ty][h] + sO[tx][h], 0.0f), sW[h], acc);
    }
  }

  const int i = i0 + ty;
  const int j = j0 + tx;
  out[(size_t)i * NN + j] = (i == j) ? 0.0f : (acc + b2[0]);
}

// ---------------------------------------------------------------------------
extern "C" void kernel_launch(void* const* d_in, const int* in_sizes, int n_in,
                              void* d_out, int out_size, void* d_ws, size_t ws_size,
                              hipStream_t stream) {
  (void)in_sizes; (void)n_in; (void)out_size; (void)ws_size;

  const float* encSub = (const float*)d_in[0];  // [N,T,D]
  const float* encObj = (const float*)d_in[1];  // [N,T,D]
  const float* W1     = (const float*)d_in[2];  // [H,2D]
  const float* b1     = (const float*)d_in[3];  // [H]
  const float* W2     = (const float*)d_in[4];  // [1,H]
  const float* b2     = (const float*)d_in[5];  // [1]
  float*       out    = (float*)d_out;          // [N,N]

  // workspace: sub | obj | s_part(+b1) | o_part  (each N*1024 floats = 1 MB)
  float* wsf    = (float*)d_ws;
  float* sub    = wsf + 0 * (size_t)NN * DD;
  float* obj    = wsf + 1 * (size_t)NN * DD;
  float* sPartB = wsf + 2 * (size_t)NN * DD;
  float* oPart  = wsf + 3 * (size_t)NN * DD;

  // K1: max-pool over tokens (HBM-bound)
  ppn_maxpool_kernel<<<dim3(1, NN, 2), dim3(256), 0, stream>>>(
      encSub, encObj, sub, obj);

  // K2: WMMA GEMMs. 2048 tiles total, 4 waves/block -> 512 blocks of 128.
  ppn_gemm_wmma_kernel<<<dim3(512), dim3(128), 0, stream>>>(
      sub, obj, W1, b1, sPartB, oPart);

  // K3: fused relu-dot pair scoring, 16x16 output tile per block.
  ppn_score_kernel<<<dim3(NN / 16, NN / 16), dim3(256), 0, stream>>>(
      sPartB, oPart, W2, b2, out);
}